// DecoderWithNMS_43267500540542
// MI455X (gfx1250) — compile-verified
//
#include <hip/hip_runtime.h>
#include <stdint.h>

// ---------------------------------------------------------------------------
// DecoderWithNMS for gfx1250 (MI455X).
// No GEMM structure -> no WMMA; CDNA5 features used instead:
//   * TENSOR_LOAD_TO_LDS (TDM, D# descriptor, TENSORcnt) for bulk LDS staging
//   * global_load_async_to_lds_b128 + s_wait_asynccnt (ASYNCcnt DMA path)
//   * wave32 ballot for suppression-bitmask generation
//   * 320KB WGP LDS keeps the 512-box table + NMS bitmask resident
// ---------------------------------------------------------------------------

#define HDIM 512
#define WDIM 512
#define BDIM 8
#define HW   (HDIM * WDIM)
#define KSEL 512
#define NBIN 4096
#define CANDMAX 4096
#define EPSF 1e-8f

// workspace layout in 32-bit units (total 278672 u32 = ~1.07 MB)
#define HIST_OFF 0u       // u32 [8*4096]
#define CNT_OFF  32768u   // u32 [8]
#define T_OFF    32776u   // u32 [8]
#define KEEP_OFF 32784u   // u32 [8*16]
#define CAND_OFF 32912u   // u64 [8*4096]   (byte off 131648, 16B aligned)
#define BOX8_OFF 98448u   // f32 [8*512*8]
#define META_OFF 131216u  // f32 [8*512*20] (byte off 524864, 16B aligned)
#define MASK_OFF 213136u  // u32 [8*512*16] (byte off 852544, 16B aligned)

typedef uint32_t u32x4 __attribute__((ext_vector_type(4)));
typedef uint32_t u32x8 __attribute__((ext_vector_type(8)));

// ------------------------------ helpers ------------------------------------

__device__ __forceinline__ uint32_t sortable(float v) {
  uint32_t u = __float_as_uint(v);
  return (u & 0x80000000u) ? ~u : (u | 0x80000000u);
}

__device__ __forceinline__ uint32_t lds_off32(const void* p) {
  // generic shared-space pointer: addr[31:0] is the LDS byte offset (ISA 10.2)
  return (uint32_t)(uintptr_t)p;
}

__device__ __forceinline__ void async_copy_b128(uint32_t ldsoff, const void* g) {
  asm volatile("global_load_async_to_lds_b128 %0, %1, off"
               :: "v"(ldsoff), "v"((unsigned long long)(uintptr_t)g)
               : "memory");
}

__device__ __forceinline__ void wait_async0() {
  asm volatile("s_wait_asynccnt 0x0" ::: "memory");
}

// Tensor Data Mover: 1-D tile of `nwords` dwords (<= 65535) global -> LDS.
// D# per CDNA5 ISA ch.8: 2 SGPR groups (2-D tensor), data_size=4B,
// tensor_dim0=tile_dim0=nwords, tensor_dim1=tile_dim1=1, stride0=nwords.
__device__ __forceinline__ void tdm_load_1d(uint32_t ldsoff, const void* g,
                                            uint32_t nwords) {
  uint64_t ga = (uint64_t)(uintptr_t)g;
  u32x4 g0;
  g0[0] = 1u;                                  // count=1, user descriptor
  g0[1] = ldsoff;                              // lds_addr
  g0[2] = (uint32_t)ga;                        // global_addr[31:0]
  g0[3] = (uint32_t)(ga >> 32) | (2u << 30);   // global_addr[56:32] | type=2
  u32x8 g1;
  g1[0] = 0x00020000u;                         // workgroup_mask=0, data_size=4B
  g1[1] = (nwords & 0xFFFFu) << 16;            // tensor_dim0[15:0] @ bits 63:48
  g1[2] = (nwords >> 16) | (1u << 16);         // tensor_dim0[31:16] | tensor_dim1=1
  g1[3] = (nwords & 0xFFFFu) << 16;            // tile_dim0 @ bits 127:112
  g1[4] = 1u;                                  // tile_dim1=1, tile_dim2=0
  g1[5] = nwords;                              // tensor_dim0_stride[31:0]
  g1[6] = 0u;                                  // stride0[47:32] | stride1[15:0]
  g1[7] = 0u;
  asm volatile("tensor_load_to_lds %0, %1" :: "s"(g0), "s"(g1) : "memory");
}

__device__ __forceinline__ void wait_tensor0() {
  __builtin_amdgcn_s_wait_tensorcnt(0);
}

__device__ __forceinline__ unsigned ballot32(bool p) {
#if __has_builtin(__builtin_amdgcn_ballot_w32)
  return __builtin_amdgcn_ballot_w32(p);
#else
  return (unsigned)__ballot(p);
#endif
}

__device__ __forceinline__ float sigmoidf_(float x) {
  return 1.0f / (1.0f + expf(-x));
}

// ------------------------- pairwise rotated IoU3D --------------------------
// A,Bm: 20-float meta in LDS: {x,y,l,w,cos,sin, cx[4], cy[4], zlo,zhi,vol, pad3}
__device__ float pair_iou3d(const float* A, const float* Bm) {
  float cax[4], cay[4], cbx[4], cby[4];
#pragma unroll
  for (int k = 0; k < 4; ++k) {
    cax[k] = A[6 + k];  cay[k] = A[10 + k];
    cbx[k] = Bm[6 + k]; cby[k] = Bm[10 + k];
  }
  float px[24], py[24];
  bool  vv[24];
  // 16 edge-edge intersections
#pragma unroll
  for (int e = 0; e < 4; ++e) {
    float a1x = cax[e], a1y = cay[e];
    float rx = cax[(e + 1) & 3] - a1x, ry = cay[(e + 1) & 3] - a1y;
#pragma unroll
    for (int f = 0; f < 4; ++f) {
      float b1x = cbx[f], b1y = cby[f];
      float sx = cbx[(f + 1) & 3] - b1x, sy = cby[(f + 1) & 3] - b1y;
      float den = rx * sy - ry * sx;
      bool  dn  = fabsf(den) > EPSF;
      float inv = 1.0f / (dn ? den : 1.0f);
      float qx = b1x - a1x, qy = b1y - a1y;
      float tt = (qx * sy - qy * sx) * inv;
      float uu = (qx * ry - qy * rx) * inv;
      bool  ok = dn && tt >= 0.0f && tt <= 1.0f && uu >= 0.0f && uu <= 1.0f;
      int kk = e * 4 + f;
      px[kk] = a1x + tt * rx; py[kk] = a1y + tt * ry; vv[kk] = ok;
    }
  }
  // corners of A inside B
  {
    float bx = Bm[0], by = Bm[1], bl = Bm[2], bw = Bm[3], bc = Bm[4], bs = Bm[5];
#pragma unroll
    for (int k = 0; k < 4; ++k) {
      float dx = cax[k] - bx, dy = cay[k] - by;
      float lx = dx * bc + dy * bs, ly = -dx * bs + dy * bc;
      vv[16 + k] = (fabsf(lx) <= bl * 0.5f + 1e-6f) && (fabsf(ly) <= bw * 0.5f + 1e-6f);
      px[16 + k] = cax[k]; py[16 + k] = cay[k];
    }
  }
  // corners of B inside A
  {
    float ax = A[0], ay = A[1], al = A[2], aw = A[3], ac = A[4], asn = A[5];
#pragma unroll
    for (int k = 0; k < 4; ++k) {
      float dx = cbx[k] - ax, dy = cby[k] - ay;
      float lx = dx * ac + dy * asn, ly = -dx * asn + dy * ac;
      vv[20 + k] = (fabsf(lx) <= al * 0.5f + 1e-6f) && (fabsf(ly) <= aw * 0.5f + 1e-6f);
      px[20 + k] = cbx[k]; py[20 + k] = cby[k];
    }
  }
  // centroid of valid points
  float n = 0.f, sx = 0.f, sy = 0.f;
#pragma unroll
  for (int k = 0; k < 24; ++k) {
    float f = vv[k] ? 1.f : 0.f;
    n += f; sx += px[k] * f; sy += py[k] * f;
  }
  float invn = 1.0f / fmaxf(n, 1.0f);
  float cx = sx * invn, cy = sy * invn;
  // fill = first valid point (argmax of bool), default element 0
  float fx = px[0], fy = py[0]; bool found = false;
#pragma unroll
  for (int k = 0; k < 24; ++k) {
    bool take = vv[k] && !found;
    fx = take ? px[k] : fx;
    fy = take ? py[k] : fy;
    found = found || vv[k];
  }
  // replace invalid, compute angles (reproduces stable argsort via rank count)
  float ang[24];
#pragma unroll
  for (int k = 0; k < 24; ++k) {
    float ppx = vv[k] ? px[k] : fx;
    float ppy = vv[k] ? py[k] : fy;
    ang[k] = atan2f(ppy - cy, ppx - cx);
    px[k] = ppx - cx; py[k] = ppy - cy;
  }
  int rnk[24];
#pragma unroll
  for (int k = 0; k < 24; ++k) {
    int r = 0;
#pragma unroll
    for (int j = 0; j < 24; ++j)
      r += ((ang[j] < ang[k]) || (ang[j] == ang[k] && j < k)) ? 1 : 0;
    rnk[k] = r;
  }
  // shoelace over angular order (successor = rank+1 mod 24)
  float a2 = 0.f;
#pragma unroll
  for (int k = 0; k < 24; ++k) {
    int want = rnk[k] + 1; want = (want == 24) ? 0 : want;
#pragma unroll
    for (int j = 0; j < 24; ++j)
      if (rnk[j] == want) a2 += px[k] * py[j] - py[k] * px[j];
  }
  float area = 0.5f * fabsf(a2);
  area = (n >= 3.0f) ? area : 0.0f;
  float z1 = fmaxf(A[14], Bm[14]);
  float z2 = fminf(A[15], Bm[15]);
  float inter = area * fmaxf(z2 - z1, 0.0f);
  return inter / (A[16] + Bm[16] - inter + EPSF);
}

// ------------------------------- kernels -----------------------------------

__global__ void k0_init(uint32_t* ws) {
  uint32_t i = blockIdx.x * blockDim.x + threadIdx.x;
  if (i < CNT_OFF + 8u) ws[i] = 0u;  // hist + counters
}

__global__ __launch_bounds__(256) void k1_hist(const float* __restrict__ src,
                                               uint32_t* __restrict__ ws) {
  __shared__ uint32_t h[NBIN];
  int b = blockIdx.y, t = threadIdx.x;
  for (int k = t; k < NBIN; k += 256) h[k] = 0u;
  __syncthreads();
  const float* sc = src + (size_t)b * 9 * HW;  // channel 0
  int base = blockIdx.x * 4096;
#pragma unroll
  for (int k = 0; k < 16; ++k) {
    int i = base + t + k * 256;
    atomicAdd(&h[sortable(sc[i]) >> 20], 1u);
  }
  __syncthreads();
  uint32_t* gh = ws + HIST_OFF + (size_t)b * NBIN;
  for (int k = t; k < NBIN; k += 256)
    if (h[k]) atomicAdd(&gh[k], h[k]);
}

__global__ void k2_thresh(uint32_t* ws) {
  int b = blockIdx.x;
  if (threadIdx.x == 0) {
    const uint32_t* gh = ws + HIST_OFF + (size_t)b * NBIN;
    uint32_t cum = 0, T = 0;
    for (int bin = NBIN - 1; bin >= 0; --bin) {
      cum += gh[bin];
      if (cum >= KSEL) { T = (uint32_t)bin; break; }
    }
    ws[T_OFF + b] = T;
  }
}

__global__ __launch_bounds__(256) void k3_compact(const float* __restrict__ src,
                                                  uint32_t* __restrict__ ws) {
  int b = blockIdx.y, t = threadIdx.x;
  const float* sc = src + (size_t)b * 9 * HW;
  uint32_t T = ws[T_OFF + b];
  uint64_t* cand = (uint64_t*)(ws + CAND_OFF) + (size_t)b * CANDMAX;
  int base = blockIdx.x * 4096;
#pragma unroll
  for (int k = 0; k < 16; ++k) {
    int i = base + t + k * 256;
    uint32_t key = sortable(sc[i]);
    if ((key >> 20) >= T) {
      uint32_t pos = atomicAdd(&ws[CNT_OFF + b], 1u);
      if (pos < CANDMAX)
        cand[pos] = ((uint64_t)key << 32) | (uint32_t)(~(uint32_t)i);
    }
  }
}

__global__ __launch_bounds__(512) void k4_select_decode(const float* __restrict__ src,
                                                        uint32_t* __restrict__ ws) {
  __shared__ uint64_t keys[CANDMAX];  // 32 KB
  int b = blockIdx.x, t = threadIdx.x;
  const uint64_t* cand = (const uint64_t*)(ws + CAND_OFF) + (size_t)b * CANDMAX;
  // async-DMA candidate table into LDS (32 KB, contiguous, b128 per lane)
  uint32_t lb = lds_off32(keys);
#pragma unroll
  for (int q = 0; q < 4; ++q) {
    uint32_t off = (uint32_t)(t + q * 512) * 16u;
    async_copy_b128(lb + off, (const char*)cand + off);
  }
  wait_async0();
  __syncthreads();
  uint32_t cnt = ws[CNT_OFF + b];
  if (cnt > CANDMAX) cnt = CANDMAX;
  for (uint32_t k = t; k < CANDMAX; k += 512)
    if (k >= cnt) keys[k] = 0ull;
  __syncthreads();
  // bitonic sort, descending (key = {sortable_conf, ~idx} -> ties: lower idx first)
  for (uint32_t kk = 2; kk <= CANDMAX; kk <<= 1) {
    for (uint32_t jj = kk >> 1; jj > 0; jj >>= 1) {
      for (uint32_t idx = t; idx < CANDMAX; idx += 512) {
        uint32_t ixj = idx ^ jj;
        if (ixj > idx) {
          uint64_t a = keys[idx], c = keys[ixj];
          bool up = ((idx & kk) == 0);
          if (up ? (a < c) : (a > c)) { keys[idx] = c; keys[ixj] = a; }
        }
      }
      __syncthreads();
    }
  }
  // every thread t < 512 decodes rank-t box
  uint64_t key = keys[t];
  uint32_t i = ~(uint32_t)(key & 0xFFFFFFFFull);
  const float* ch = src + (size_t)b * 9 * HW;
  float o0 = ch[i],          o1 = ch[HW + i],     o2 = ch[2 * HW + i];
  float o3 = ch[3 * HW + i], o4 = ch[4 * HW + i], o5 = ch[5 * HW + i];
  float o6 = ch[6 * HW + i], o7 = ch[7 * HW + i], o8 = ch[8 * HW + i];
  float conf = sigmoidf_(o0);
  float x = sigmoidf_(o1) + (float)(i % WDIM);
  float y = sigmoidf_(o2) + (float)(i / WDIM);
  float z = sigmoidf_(o3) * 4.0f;
  float l = expf(o4) * 3.9f, w = expf(o5) * 1.6f, hh = expf(o6) * 1.56f;
  float yaw = atan2f(tanhf(o7), tanhf(o8));
  float* bx8 = (float*)ws + BOX8_OFF + ((size_t)b * KSEL + t) * 8;
  bx8[0] = conf; bx8[1] = x; bx8[2] = y; bx8[3] = z;
  bx8[4] = l;    bx8[5] = w; bx8[6] = hh; bx8[7] = yaw;
  float c = cosf(yaw), s = sinf(yaw);
  float* m = (float*)ws + META_OFF + ((size_t)b * KSEL + t) * 20;
  m[0] = x; m[1] = y; m[2] = l; m[3] = w; m[4] = c; m[5] = s;
  const float LX[4] = {0.5f, 0.5f, -0.5f, -0.5f};
  const float LY[4] = {0.5f, -0.5f, -0.5f, 0.5f};
#pragma unroll
  for (int k = 0; k < 4; ++k) {
    float lx = LX[k] * l, ly = LY[k] * w;
    m[6 + k]  = x + lx * c - ly * s;
    m[10 + k] = y + lx * s + ly * c;
  }
  m[14] = z - hh * 0.5f; m[15] = z + hh * 0.5f; m[16] = l * w * hh;
  m[17] = 0.f; m[18] = 0.f; m[19] = 0.f;
  // keep0 = conf > VALIDCONF, one ballot word per wave32
  unsigned kw = ballot32(conf > 0.5f);
  if ((t & 31) == 0) ws[KEEP_OFF + b * 16 + (t >> 5)] = kw;
}

__global__ __launch_bounds__(256) void k5_iou_mask(uint32_t* __restrict__ ws) {
  __shared__ float sm[KSEL * 20];  // 40 KB box-meta table, TDM-staged
  int b = blockIdx.y, t = threadIdx.x;
  const float* gmeta = (const float*)ws + META_OFF + (size_t)b * KSEL * 20;
  if (t < 32) {  // wave 0 drives the Tensor Data Mover, waits on TENSORcnt
    tdm_load_1d(lds_off32(sm), gmeta, KSEL * 20);
    wait_tensor0();
  }
  __syncthreads();
  int wave = t >> 5, lane = t & 31;
  int i0 = blockIdx.x * 16;
#pragma unroll 1
  for (int r = 0; r < 16; ++r) {
    int i = i0 + r;
    const float* A = &sm[i * 20];
#pragma unroll 1
    for (int wwi = 0; wwi < 2; ++wwi) {
      int word = wave * 2 + wwi;
      int j = lane + word * 32;
      float iou = pair_iou3d(A, &sm[j * 20]);
      bool bit = (iou > 0.1f) && (j > i);
      unsigned mw = ballot32(bit);
      if (lane == 0)
        ws[MASK_OFF + ((size_t)b * KSEL + i) * 16 + word] = mw;
    }
  }
}

__global__ __launch_bounds__(256) void k6_suppress(uint32_t* __restrict__ ws,
                                                   float* __restrict__ outp) {
  __shared__ uint32_t smask[KSEL * 16];  // 32 KB suppression bitmask, TDM-staged
  __shared__ uint32_t skeep[16];
  int b = blockIdx.x, t = threadIdx.x;
  const uint32_t* gm = ws + MASK_OFF + (size_t)b * KSEL * 16;
  if (t < 16) skeep[t] = ws[KEEP_OFF + b * 16 + t];
  if (t < 32) {  // wave 0 drives the Tensor Data Mover, waits on TENSORcnt
    tdm_load_1d(lds_off32(smask), gm, KSEL * 16);
    wait_tensor0();
  }
  __syncthreads();
  // greedy NMS: 512 steps over 16 words, all in LDS
  for (int i = 0; i < KSEL; ++i) {
    bool ki = (skeep[i >> 5] >> (i & 31)) & 1u;
    __syncthreads();
    if (ki && t < 16) skeep[t] &= ~smask[i * 16 + t];
    __syncthreads();
  }
  const float* bx8 = (const float*)ws + BOX8_OFF + (size_t)b * KSEL * 8;
  for (int idx = t; idx < KSEL * 8; idx += 256) {
    int k = idx >> 3;
    bool kb = (skeep[k >> 5] >> (k & 31)) & 1u;
    outp[(size_t)b * KSEL * 8 + idx] = kb ? bx8[idx] : 0.0f;
  }
}

// ------------------------------- launcher ----------------------------------

extern "C" void kernel_launch(void* const* d_in, const int* in_sizes, int n_in,
                              void* d_out, int out_size, void* d_ws, size_t ws_size,
                              hipStream_t stream) {
  (void)in_sizes; (void)n_in; (void)out_size; (void)ws_size;
  const float* src = (const float*)d_in[0];  // (8, 9, 512, 512) fp32; target unused
  float*    outp = (float*)d_out;            // (8, 512, 8) fp32
  uint32_t* ws   = (uint32_t*)d_ws;          // needs ~1.07 MB

  k0_init<<<(CNT_OFF + 8u + 255u) / 256u, 256, 0, stream>>>(ws);
  k1_hist<<<dim3(HW / 4096, BDIM), 256, 0, stream>>>(src, ws);
  k2_thresh<<<BDIM, 32, 0, stream>>>(ws);
  k3_compact<<<dim3(HW / 4096, BDIM), 256, 0, stream>>>(src, ws);
  k4_select_decode<<<BDIM, 512, 0, stream>>>(src, ws);
  k5_iou_mask<<<dim3(KSEL / 16, BDIM), 256, 0, stream>>>(ws);
  k6_suppress<<<BDIM, 256, 0, stream>>>(ws, outp);
}